// MyNet_66365834658260
// MI455X (gfx1250) — compile-verified
//
#include <hip/hip_runtime.h>

typedef float v2f __attribute__((ext_vector_type(2)));
typedef float v8f __attribute__((ext_vector_type(8)));

#define N_NODES 100000
#define N_EDGES 3200000
#define F_IN    512
#define H_OUT   16
#define N_TILES (N_NODES / 16)   // 6250, exact

// ---------------- zero-init workspace (acc + deg) ----------------
__global__ void gcn_init_kernel(float* __restrict__ acc, unsigned* __restrict__ deg) {
    int i = blockIdx.x * blockDim.x + threadIdx.x;
    if (i < N_NODES * H_OUT) acc[i] = 0.0f;
    if (i < N_NODES)         deg[i] = 0u;
}

// ---------------- h = x @ W1 via V_WMMA_F32_16X16X4_F32 ----------------
__global__ __launch_bounds__(256) void gcn_gemm_kernel(const float* __restrict__ x,
                                                       const float* __restrict__ W,
                                                       float* __restrict__ h) {
    __shared__ float lW[F_IN * H_OUT];  // 32 KB: whole weight matrix
    for (int idx = threadIdx.x; idx < (F_IN * H_OUT) / 4; idx += 256)
        ((float4*)lW)[idx] = ((const float4*)W)[idx];
    __syncthreads();

    const int wave = threadIdx.x >> 5;
    const int lane = threadIdx.x & 31;
    const int tile = blockIdx.x * 8 + wave;
    if (tile >= N_TILES) return;          // wave-uniform: EXEC stays all-ones

    const int half = lane >> 4;           // 0: K+{0,1}, 1: K+{2,3}
    const int m    = lane & 15;

    // A fragment source: row (tile*16+m), starting at K offset 2*half (8B aligned)
    const float* xrow = x + (size_t)(tile * 16 + m) * F_IN + 2 * half;
    // B fragment source: W[k + 2*half + {0,1}][m]
    const float* wcol = lW + 2 * half * H_OUT + m;

    v8f c = {};
    #pragma unroll 4
    for (int k = 0; k < F_IN; k += 4) {
        float2 av = *(const float2*)(xrow + k);
        v2f a; a[0] = av.x; a[1] = av.y;                 // VGPR0=K(+0/+2), VGPR1=K(+1/+3)
        v2f b; b[0] = wcol[k * H_OUT];                   // W[k+2*half  ][m]
               b[1] = wcol[k * H_OUT + H_OUT];           // W[k+2*half+1][m]
        c = __builtin_amdgcn_wmma_f32_16x16x4_f32(
                /*neg_a=*/false, a, /*neg_b=*/false, b,
                /*c_mod=*/(short)0, c, /*reuse_a=*/false, /*reuse_b=*/false);
    }

    // C/D layout: VGPR r -> M = r + 8*half, N = lane&15
    float* hrow = h + (size_t)(tile * 16 + half * 8) * H_OUT + m;
    #pragma unroll
    for (int r = 0; r < 8; ++r) hrow[r * H_OUT] = c[r];
}

// ---------------- in-degree via u32 atomics ----------------
__global__ void gcn_degree_kernel(const long long* __restrict__ dst,
                                  unsigned* __restrict__ deg) {
    int e = blockIdx.x * blockDim.x + threadIdx.x;
    if (e < N_EDGES) atomicAdd(&deg[(int)dst[e]], 1u);
}

// ---------------- dinv = rsqrt(deg + 1)  (self-loop adds 1) ----------------
__global__ void gcn_dinv_kernel(const unsigned* __restrict__ deg,
                                float* __restrict__ dinv) {
    int i = blockIdx.x * blockDim.x + threadIdx.x;
    if (i < N_NODES) dinv[i] = rsqrtf((float)(deg[i] + 1u));
}

// ---------------- message gather/scale/scatter-add ----------------
__global__ void gcn_scatter_kernel(const long long* __restrict__ src,
                                   const long long* __restrict__ dst,
                                   const float* __restrict__ h,
                                   const float* __restrict__ dinv,
                                   float* __restrict__ acc) {
    int e = blockIdx.x * blockDim.x + threadIdx.x;
    if (e >= N_EDGES) return;
    const int s = (int)src[e];
    const int d = (int)dst[e];
    const float w = dinv[s] * dinv[d];
    const float4* hs = (const float4*)(h + (size_t)s * H_OUT);
    float* ad = acc + (size_t)d * H_OUT;
    #pragma unroll
    for (int q = 0; q < 4; ++q) {
        float4 v = hs[q];
        atomicAdd(ad + 4 * q + 0, v.x * w);
        atomicAdd(ad + 4 * q + 1, v.y * w);
        atomicAdd(ad + 4 * q + 2, v.z * w);
        atomicAdd(ad + 4 * q + 3, v.w * w);
    }
}

// ---------------- self-loop + bias + relu + log_softmax ----------------
__global__ void gcn_finalize_kernel(const float* __restrict__ acc,
                                    const float* __restrict__ h,
                                    const float* __restrict__ dinv,
                                    const float* __restrict__ bias,
                                    float* __restrict__ out) {
    int i = blockIdx.x * blockDim.x + threadIdx.x;
    if (i >= N_NODES) return;
    const float di = dinv[i];
    const float wself = di * di;
    float v[H_OUT];
    float mx = -3.402823466e38f;
    const float4* a4 = (const float4*)(acc + (size_t)i * H_OUT);
    const float4* h4 = (const float4*)(h + (size_t)i * H_OUT);
    #pragma unroll
    for (int q = 0; q < 4; ++q) {
        float4 a = a4[q], hv = h4[q];
        v[4*q+0] = a.x + hv.x * wself + bias[4*q+0];
        v[4*q+1] = a.y + hv.y * wself + bias[4*q+1];
        v[4*q+2] = a.z + hv.z * wself + bias[4*q+2];
        v[4*q+3] = a.w + hv.w * wself + bias[4*q+3];
    }
    #pragma unroll
    for (int f = 0; f < H_OUT; ++f) {
        v[f] = fmaxf(v[f], 0.0f);          // relu
        mx = fmaxf(mx, v[f]);
    }
    float ssum = 0.0f;
    #pragma unroll
    for (int f = 0; f < H_OUT; ++f) ssum += __expf(v[f] - mx);
    const float lse = __logf(ssum) + mx;
    #pragma unroll
    for (int f = 0; f < H_OUT; ++f) out[(size_t)i * H_OUT + f] = v[f] - lse;
}

extern "C" void kernel_launch(void* const* d_in, const int* in_sizes, int n_in,
                              void* d_out, int out_size, void* d_ws, size_t ws_size,
                              hipStream_t stream) {
    const float*     x    = (const float*)d_in[0];
    const long long* ei   = (const long long*)d_in[1];   // int64 [2, E]: row0=src, row1=dst
    const float*     W1   = (const float*)d_in[2];
    const float*     b1   = (const float*)d_in[3];
    float*           out  = (float*)d_out;

    char* ws = (char*)d_ws;
    float*    h    = (float*)ws;                                         // 6.4 MB
    float*    acc  = (float*)(ws + (size_t)N_NODES * H_OUT * 4);         // 6.4 MB
    unsigned* deg  = (unsigned*)(ws + 2 * (size_t)N_NODES * H_OUT * 4);  // 400 KB
    float*    dinv = (float*)(ws + 2 * (size_t)N_NODES * H_OUT * 4 + (size_t)N_NODES * 4);

    const long long* src = ei;
    const long long* dst = ei + N_EDGES;

    gcn_init_kernel<<<(N_NODES * H_OUT + 255) / 256, 256, 0, stream>>>(acc, deg);
    gcn_gemm_kernel<<<(N_TILES + 7) / 8, 256, 0, stream>>>(x, W1, h);
    gcn_degree_kernel<<<N_EDGES / 256, 256, 0, stream>>>(dst, deg);
    gcn_dinv_kernel<<<(N_NODES + 255) / 256, 256, 0, stream>>>(deg, dinv);
    gcn_scatter_kernel<<<N_EDGES / 256, 256, 0, stream>>>(src, dst, h, dinv, acc);
    gcn_finalize_kernel<<<(N_NODES + 255) / 256, 256, 0, stream>>>(acc, h, dinv, b1, out);
}